// GraphConvolution_14173392077176
// MI455X (gfx1250) — compile-verified
//
#include <hip/hip_runtime.h>
#include <hip/hip_bf16.h>
#include <math.h>

typedef __attribute__((ext_vector_type(2))) float v2f;
typedef __attribute__((ext_vector_type(8))) float v8f;

#define D 64
#define TWO_D 128

// ---------------------------------------------------------------------------
// Kernel 1: edge-weighted scatter-add (SpMM).  One wave32 per edge; lane L
// handles features L and L+32.  Native f32 atomic adds -> L2-side RMW; the
// 25.6MB h_agg working set is resident in MI455X's 192MB L2.
// ---------------------------------------------------------------------------
__global__ void gcn_scatter_kernel(const float* __restrict__ h,
                                   const float* __restrict__ edge_w,
                                   const int* __restrict__ edge_src,
                                   const int* __restrict__ edge_dst,
                                   float* __restrict__ h_agg,
                                   int E) {
    int tid  = blockIdx.x * blockDim.x + threadIdx.x;
    int edge = tid >> 5;
    int lane = tid & 31;
    if (edge >= E) return;

    int src  = edge_src[edge];
    int dst  = edge_dst[edge];
    float w  = edge_w[edge];

    const float* hs = h + (size_t)src * D;
    float* ha       = h_agg + (size_t)dst * D;

    float v0 = hs[lane]      * w;
    float v1 = hs[lane + 32] * w;
    unsafeAtomicAdd(&ha[lane],      v0);
    unsafeAtomicAdd(&ha[lane + 32], v1);
}

// ---------------------------------------------------------------------------
// Kernel 2: fused  out = theta*([h_agg|i] @ W) + (1-theta)*r + i
// using V_WMMA_F32_16X16X4_F32.  One wave per 16-row tile.
//
// W is staged in LDS pair-interleaved: Blds2[p][n] = {W[2p][n], W[2p+1][n]}
// so every B fragment is a single ds_load_b64 at an immediate offset.
// The k-loop is split at k=64 (h_agg half vs i half) so the A pointer is
// loop-invariant per half: each A fragment is one global_load_b64.
// ---------------------------------------------------------------------------
__global__ void __launch_bounds__(256)
gcn_gemm_kernel(const float* __restrict__ h_agg,
                const float* __restrict__ ifeat,
                const float* __restrict__ W,      // [128,64] row-major
                const float* __restrict__ lamda_p,
                const float* __restrict__ alpha_p,
                const int*   __restrict__ layer_p,
                float* __restrict__ out,
                int N) {
    __shared__ float Blds[TWO_D * D];             // 32 KB, pair-interleaved

    int t = threadIdx.x;
    // Cooperative staging with pair interleave:
    //   dst index = (k>>1)*128 + n*2 + (k&1)
    #pragma unroll 4
    for (int idx = t; idx < TWO_D * D; idx += 256) {
        int k = idx >> 6;                         // 0..127
        int n = idx & 63;
        Blds[((k >> 1) << 7) + (n << 1) + (k & 1)] = W[idx];
    }
    __syncthreads();

    int wave = t >> 5;                            // 0..7
    int lane = t & 31;
    int rowTile = blockIdx.x * 8 + wave;          // 16 rows per wave tile
    int numTiles = (N + 15) >> 4;
    if (rowTile >= numTiles) return;              // wave-uniform: EXEC all-1s

    int m        = lane & 15;                     // row/col within tile
    int lanehalf = lane >> 4;                     // 0 or 1
    int khalf    = lanehalf << 1;                 // lanes 0-15 -> K{0,1}; 16-31 -> K{2,3}
    int row0     = rowTile * 16;
    int arow     = row0 + m;                      // A-matrix row owned by this lane

    float lamda = lamda_p[0];
    float alpha = alpha_p[0];
    float theta = fminf(1.0f, logf(lamda / (float)layer_p[0] + 1.0f));

    // A bases (8B-aligned: khalf*4 bytes is a multiple of 8)
    const float* Ah = h_agg + (size_t)arow * D + khalf;   // support cols 0..63
    const float* Ai = ifeat + (size_t)arow * D + khalf;   // support cols 64..127

    // B base in pair units: element (p, n) at Bp[p*64 + n]; lane-dependent part
    // folded into the base so all loop offsets are immediates.
    const v2f* Bp = ((const v2f*)Blds) + (lanehalf << 6) + m;

    v8f acc0 = {}, acc1 = {}, acc2 = {}, acc3 = {};

    #pragma unroll
    for (int kk = 0; kk < 16; ++kk) {             // K = 0..63  (h_agg half)
        v2f a = *(const v2f*)(Ah + kk * 4);
        int pb = kk << 7;                         // pair stride: 2 pairs/step * 64
        acc0 = __builtin_amdgcn_wmma_f32_16x16x4_f32(false, a, false, Bp[pb +  0], (short)0, acc0, false, false);
        acc1 = __builtin_amdgcn_wmma_f32_16x16x4_f32(false, a, false, Bp[pb + 16], (short)0, acc1, false, false);
        acc2 = __builtin_amdgcn_wmma_f32_16x16x4_f32(false, a, false, Bp[pb + 32], (short)0, acc2, false, false);
        acc3 = __builtin_amdgcn_wmma_f32_16x16x4_f32(false, a, false, Bp[pb + 48], (short)0, acc3, false, false);
    }
    #pragma unroll
    for (int kk = 16; kk < 32; ++kk) {            // K = 64..127 (i half)
        v2f a = *(const v2f*)(Ai + (kk - 16) * 4);
        int pb = kk << 7;
        acc0 = __builtin_amdgcn_wmma_f32_16x16x4_f32(false, a, false, Bp[pb +  0], (short)0, acc0, false, false);
        acc1 = __builtin_amdgcn_wmma_f32_16x16x4_f32(false, a, false, Bp[pb + 16], (short)0, acc1, false, false);
        acc2 = __builtin_amdgcn_wmma_f32_16x16x4_f32(false, a, false, Bp[pb + 32], (short)0, acc2, false, false);
        acc3 = __builtin_amdgcn_wmma_f32_16x16x4_f32(false, a, false, Bp[pb + 48], (short)0, acc3, false, false);
    }

    // Epilogue.  C/D layout: VGPR v holds row v (lanes 0-15) / row v+8 (16-31),
    // column = (lane&15) + 16*nt.
    float one_m_theta = 1.0f - theta;
    float one_m_alpha = 1.0f - alpha;
    int   rbase = row0 + (lanehalf << 3);

    #pragma unroll
    for (int v = 0; v < 8; ++v) {
        int orow = rbase + v;
        if (orow >= N) break;                     // tail guard (after all WMMAs)
        const float* ha = h_agg + (size_t)orow * D;
        const float* ii = ifeat + (size_t)orow * D;
        float*       oo = out   + (size_t)orow * D;
        #pragma unroll
        for (int nt = 0; nt < 4; ++nt) {
            int c = nt * 16 + m;
            float aggv = ha[c];
            float iv   = ii[c];
            float r    = one_m_alpha * aggv + alpha * iv;
            float g    = (nt == 0) ? acc0[v] : (nt == 1) ? acc1[v]
                       : (nt == 2) ? acc2[v] : acc3[v];
            oo[c] = theta * g + one_m_theta * r + iv;
        }
    }
}

// ---------------------------------------------------------------------------
// Launch
// ---------------------------------------------------------------------------
extern "C" void kernel_launch(void* const* d_in, const int* in_sizes, int n_in,
                              void* d_out, int out_size, void* d_ws, size_t ws_size,
                              hipStream_t stream) {
    const float* h        = (const float*)d_in[0];
    const float* ifeat    = (const float*)d_in[1];
    const float* weight   = (const float*)d_in[2];
    const float* edge_w   = (const float*)d_in[3];
    const float* lamda_p  = (const float*)d_in[4];
    const float* alpha_p  = (const float*)d_in[5];
    const int*   edge_src = (const int*)d_in[6];
    const int*   edge_dst = (const int*)d_in[7];
    const int*   layer_p  = (const int*)d_in[8];
    float*       out      = (float*)d_out;

    int N = in_sizes[0] / D;
    int E = in_sizes[3];

    float* h_agg = (float*)d_ws;                  // N*64 floats (25.6 MB)

    hipMemsetAsync(h_agg, 0, (size_t)N * D * sizeof(float), stream);

    // SpMM scatter: one wave per edge
    {
        long long threads = (long long)E * 32;
        int blocks = (int)((threads + 255) / 256);
        gcn_scatter_kernel<<<blocks, 256, 0, stream>>>(h, edge_w, edge_src, edge_dst,
                                                       h_agg, E);
    }

    // Fused WMMA GEMM + epilogue: 8 row-tiles (of 16) per 256-thread block
    {
        int numTiles = (N + 15) / 16;
        int blocks = (numTiles + 7) / 8;
        gcn_gemm_kernel<<<blocks, 256, 0, stream>>>(h_agg, ifeat, weight,
                                                    lamda_p, alpha_p, layer_p,
                                                    out, N);
    }
}